// KalmanFilter_49065706390192
// MI455X (gfx1250) — compile-verified
//
#include <hip/hip_runtime.h>

#define BB 128
#define TT 256
#define SS 64
#define MM 8
#define LD 68   // padded row stride for 64x64 matrices (bank-conflict-free b64 ops)
#define LDT 12  // padded row stride for 64x8 (HPT / XT) matrices

typedef __attribute__((ext_vector_type(2))) float v2f;
typedef __attribute__((ext_vector_type(8))) float v8f;

static __device__ __forceinline__ v8f wmma4(v2f a, v2f b, v8f c) {
  // D = A(16x4) * B(4x16) + C(16x16), all f32
  return __builtin_amdgcn_wmma_f32_16x16x4_f32(false, a, false, b, (short)0, c,
                                               false, false);
}

__global__ __launch_bounds__(256, 1) void kalman_wmma_kernel(
    const float* __restrict__ gy, const float* __restrict__ gF,
    const float* __restrict__ gH, const float* __restrict__ gQ,
    const float* __restrict__ gR, const float* __restrict__ gm0,
    const float* __restrict__ gP0, float* __restrict__ gout) {
  __shared__ float sF[SS * LD];
  __shared__ float sP[SS * LD];
  __shared__ float sW[SS * LD];
  __shared__ float sH[MM * SS];
  __shared__ float sHPT[SS * LDT];  // (H P)^T, 64 x 8 (cols 8..11 unused)
  __shared__ float sXT[SS * LDT];   // -(S^-1 H P)^T = -K, 64 x 8
  __shared__ float sSinn[MM * MM];
  __shared__ float sSinv[MM * MM];
  __shared__ float sAug[MM * 18];
  __shared__ float sM[SS];
  __shared__ float sMpost[SS];
  __shared__ float sResid[MM];

  const int b = blockIdx.x;
  const int tid = (int)threadIdx.x;
  const int lane = tid & 31;
  const int wave = tid >> 5;
  const int r = lane & 15;   // row/col within 16-wide tile
  const int hi = lane >> 4;  // lane-half select
  const int koff = 2 * hi;   // K offset inside a K=4 slab

  // ---- stage constants and initial state ----
  for (int i = tid; i < SS * SS; i += 256) sF[(i >> 6) * LD + (i & 63)] = gF[i];
  for (int i = tid; i < MM * SS; i += 256) sH[i] = gH[i];
  for (int i = tid; i < SS * SS; i += 256)
    sP[(i >> 6) * LD + (i & 63)] = gP0[(size_t)b * SS * SS + i];
  if (tid < SS) sM[tid] = gm0[b * SS + tid];

  // loop-invariant per-thread addends (Q for the P_new tiles, R for Sinn)
  float qreg[2][8];
  for (int tt = 0; tt < 2; ++tt) {
    int tile = wave + 8 * tt;
    int ti = tile >> 2, tj = tile & 3;
    for (int v = 0; v < 8; ++v)
      qreg[tt][v] = gQ[(ti * 16 + v + 8 * hi) * SS + tj * 16 + r];
  }
  float rreg = (tid < MM * MM) ? gR[tid] : 0.0f;

  float* out_means = gout;                          // (B,T,M)
  float* out_covs = gout + (size_t)BB * TT * MM;    // (B,T,M,M)

  __syncthreads();

  for (int t = 0; t < TT; ++t) {
    // ---- phase 1: HPT = P * H^T  (64x16, cols >= 8 are zero) -------------
    // A = rows of P (contiguous), B = H^T (row reads of H, zero for r>=8).
    {
      int ti = wave & 3;  // waves 4..7 duplicate (identical writes, no divergence)
      v8f acc = {};
      const float* ab = &sP[(ti * 16 + r) * LD + koff];
#pragma unroll
      for (int kk = 0; kk < SS; kk += 4) {
        v2f a = *(const v2f*)(ab + kk);
        v2f bv;
        if (r < MM) {
          bv = *(const v2f*)&sH[r * SS + kk + koff];
        } else {
          bv.x = 0.0f;
          bv.y = 0.0f;
        }
        acc = wmma4(a, bv, acc);
      }
      if (r < MM) {
#pragma unroll
        for (int v = 0; v < 8; ++v)
          sHPT[(ti * 16 + v + 8 * hi) * LDT + r] = acc[v];
      }
    }
    __syncthreads();

    // ---- phase 2: Sinn = H P H^T + R (== output cov), Hm (== output mean)
    if (tid < MM * MM) {
      int ii = tid >> 3;
      float acc = rreg;
      for (int s = 0; s < SS; ++s) acc += sH[ii * SS + s] * sHPT[s * LDT + (tid & 7)];
      sSinn[tid] = acc;  // symmetric: [i][j] == [j][i]
      out_covs[((size_t)b * TT + t) * (MM * MM) + tid] = acc;
    }
    float hm = 0.0f;
    if (tid < MM) {
      for (int s = 0; s < SS; ++s) hm += sH[tid * SS + s] * sM[s];
      out_means[((size_t)b * TT + t) * MM + tid] = hm;
    }
    if (t == TT - 1) break;
    if (tid < MM) sResid[tid] = gy[((size_t)b * TT + t) * MM + tid] - hm;
    __syncthreads();

    // ---- phase 3: Sinv = Sinn^-1 (8x8 SPD, wave-synchronous GJ in wave 0)
    if (wave == 0) {
      for (int p = lane; p < 128; p += 32) {
        int i = p >> 4, j = p & 15;
        sAug[i * 18 + j] = (j < 8) ? sSinn[i * 8 + j] : ((j - 8 == i) ? 1.0f : 0.0f);
      }
      __builtin_amdgcn_wave_barrier();
      for (int k = 0; k < 8; ++k) {
        float invp = 1.0f / sAug[k * 18 + k];
        if (lane < 16) sAug[k * 18 + lane] *= invp;
        __builtin_amdgcn_wave_barrier();
        for (int p = lane; p < 128; p += 32) {
          int i = p >> 4, j = p & 15;
          float f = sAug[i * 18 + k];
          float rk = sAug[k * 18 + j];
          if (i != k) sAug[i * 18 + j] -= f * rk;
        }
        __builtin_amdgcn_wave_barrier();
      }
      for (int p = lane; p < 64; p += 32)
        sSinv[p] = sAug[(p >> 3) * 18 + 8 + (p & 7)];
    }
    __syncthreads();

    // ---- phase 4: XT[c][m] = -(Sinv * HP)[m][c] = -K  --------------------
    for (int idx = tid; idx < SS * MM; idx += 256) {
      int c = idx >> 3, m = idx & 7;
      float acc = 0.0f;
      for (int k = 0; k < 8; ++k) acc += sSinv[m * 8 + k] * sHPT[c * LDT + k];
      sXT[c * LDT + m] = -acc;
    }
    __syncthreads();

    // ---- phase 5: m_post = m + K * resid ---------------------------------
    if (tid < SS) {
      float acc = sM[tid];
      for (int k = 0; k < 8; ++k) acc -= sXT[tid * LDT + k] * sResid[k];
      sMpost[tid] = acc;
    }

    // ---- phase 6: P_post = P + (-K) * HP  (WMMA, C accumulator, in place)
    {
#pragma unroll
      for (int tt = 0; tt < 2; ++tt) {
        int tile = wave + 8 * tt;
        int ti = tile >> 2, tj = tile & 3;
        v8f c;
#pragma unroll
        for (int v = 0; v < 8; ++v)
          c[v] = sP[(ti * 16 + v + 8 * hi) * LD + tj * 16 + r];
#pragma unroll
        for (int kk = 0; kk < 8; kk += 4) {
          v2f a = *(const v2f*)&sXT[(ti * 16 + r) * LDT + kk + koff];
          v2f bv = *(const v2f*)&sHPT[(tj * 16 + r) * LDT + kk + koff];
          c = wmma4(a, bv, c);
        }
#pragma unroll
        for (int v = 0; v < 8; ++v)
          sP[(ti * 16 + v + 8 * hi) * LD + tj * 16 + r] = c[v];
      }
    }
    __syncthreads();

    // ---- phase 7: m_new = F * m_post ; W = F * P_post (WMMA) -------------
    if (tid < SS) {
      float acc = 0.0f;
      const float* frow = &sF[tid * LD];
      for (int s = 0; s < SS; ++s) acc += frow[s] * sMpost[s];
      sM[tid] = acc;
    }
    {
#pragma unroll
      for (int tt = 0; tt < 2; ++tt) {
        int tile = wave + 8 * tt;
        int ti = tile >> 2, tj = tile & 3;
        v8f acc = {};
        const float* ab = &sF[(ti * 16 + r) * LD + koff];
        const float* bb = &sP[(tj * 16 + r) * LD + koff];  // P_post symmetric
#pragma unroll
        for (int kk = 0; kk < SS; kk += 4) {
          v2f a = *(const v2f*)(ab + kk);
          v2f bv = *(const v2f*)(bb + kk);
          acc = wmma4(a, bv, acc);
        }
#pragma unroll
        for (int v = 0; v < 8; ++v)
          sW[(ti * 16 + v + 8 * hi) * LD + tj * 16 + r] = acc[v];
      }
    }
    __syncthreads();

    // ---- phase 8: P_new = W * F^T + Q (WMMA) -----------------------------
    {
#pragma unroll
      for (int tt = 0; tt < 2; ++tt) {
        int tile = wave + 8 * tt;
        int ti = tile >> 2, tj = tile & 3;
        v8f acc = {};
        const float* ab = &sW[(ti * 16 + r) * LD + koff];
        const float* bb = &sF[(tj * 16 + r) * LD + koff];  // B = F^T -> rows of F
#pragma unroll
        for (int kk = 0; kk < SS; kk += 4) {
          v2f a = *(const v2f*)(ab + kk);
          v2f bv = *(const v2f*)(bb + kk);
          acc = wmma4(a, bv, acc);
        }
#pragma unroll
        for (int v = 0; v < 8; ++v)
          sP[(ti * 16 + v + 8 * hi) * LD + tj * 16 + r] = acc[v] + qreg[tt][v];
      }
    }
    __syncthreads();

    // ---- phase 9: symmetrize P_new ---------------------------------------
    float tmp[16];
#pragma unroll
    for (int k2 = 0; k2 < 16; ++k2) {
      int idx = tid + k2 * 256;
      int ss2 = idx >> 6, cc2 = idx & 63;
      tmp[k2] = 0.5f * (sP[ss2 * LD + cc2] + sP[cc2 * LD + ss2]);
    }
    __syncthreads();
#pragma unroll
    for (int k2 = 0; k2 < 16; ++k2) {
      int idx = tid + k2 * 256;
      sP[(idx >> 6) * LD + (idx & 63)] = tmp[k2];
    }
    __syncthreads();
  }
}

extern "C" void kernel_launch(void* const* d_in, const int* in_sizes, int n_in,
                              void* d_out, int out_size, void* d_ws,
                              size_t ws_size, hipStream_t stream) {
  (void)in_sizes; (void)n_in; (void)out_size; (void)d_ws; (void)ws_size;
  const float* y = (const float*)d_in[0];
  const float* F = (const float*)d_in[1];
  const float* H = (const float*)d_in[2];
  const float* Q = (const float*)d_in[3];
  const float* R = (const float*)d_in[4];
  const float* m0 = (const float*)d_in[5];
  const float* P0 = (const float*)d_in[6];
  float* out = (float*)d_out;
  kalman_wmma_kernel<<<dim3(BB), dim3(256), 0, stream>>>(y, F, H, Q, R, m0, P0, out);
}